// GConvMultiScale_283467842539
// MI455X (gfx1250) — compile-verified
//
#include <hip/hip_runtime.h>

#define NNODES  50000
#define NEDGES  800000
#define NSCALES 4
#define DH      128
#define MTILES  (NNODES / 16)   // 3125, exact

typedef float v2f __attribute__((ext_vector_type(2)));
typedef float v4f __attribute__((ext_vector_type(4)));
typedef float v8f __attribute__((ext_vector_type(8)));

// ---------------- degree / normalization ----------------

__global__ __launch_bounds__(256) void k_deg_init(float* __restrict__ deg) {
  int i = blockIdx.x * 256 + threadIdx.x;
  if (i < NNODES) deg[i] = 1.0f;  // self-loop weight folded in
}

__global__ __launch_bounds__(256) void k_deg_accum(const int* __restrict__ dst,
                                                   const float* __restrict__ w,
                                                   float* __restrict__ deg) {
  int e = blockIdx.x * 256 + threadIdx.x;
  if (e < NEDGES) atomicAdd(&deg[dst[e]], w[e]);
}

__global__ __launch_bounds__(256) void k_dinv(float* __restrict__ deg) {
  int i = blockIdx.x * 256 + threadIdx.x;
  if (i < NNODES) {
    float d = deg[i];
    deg[i] = (d > 0.0f) ? rsqrtf(d) : 0.0f;  // in-place deg -> dinv
  }
}

__global__ __launch_bounds__(256) void k_norm(const int* __restrict__ src,
                                              const int* __restrict__ dst,
                                              const float* __restrict__ w,
                                              const float* __restrict__ dinv,
                                              float* __restrict__ norm) {
  int e = blockIdx.x * 256 + threadIdx.x;
  if (e < NEDGES) norm[e] = dinv[src[e]] * w[e] * dinv[dst[e]];
}

// ---------------- GEMM: C[M,128] = H[M,128] @ W[128,128] via V_WMMA_F32_16X16X4_F32 ----
// One wave = one 16-row strip across all 128 output cols (8 N-tiles).
// A 16x4 f32 frag: lanes 0-15 hold (M=lane, K=0..1), lanes 16-31 hold (M=lane-16, K=2..3).
// B 4x16 f32 frag: lanes 0-15 hold (K=0..1, N=lane), lanes 16-31 hold (K=2..3, N=lane-16).
// C/D 16x16 f32: VGPR i, lanes 0-15 -> (M=i, N=lane); lanes 16-31 -> (M=8+i, N=lane-16).

__global__ __launch_bounds__(256) void k_gemm(const float* __restrict__ H,
                                              const float* __restrict__ W,
                                              float* __restrict__ C) {
  int wave  = threadIdx.x >> 5;
  int lane  = threadIdx.x & 31;
  int tileM = blockIdx.x * 8 + wave;
  if (tileM >= MTILES) return;          // wave-uniform: EXEC stays all-ones
  int half = lane >> 4;                 // 0: low half-lanes, 1: high
  int li   = lane & 15;

  v8f acc[8] = {};

  const float* Arow = H + (size_t)(tileM * 16 + li) * DH + 2 * half;

  for (int k = 0; k < DH; k += 4) {
    v2f a = *(const v2f*)(Arow + k);
#pragma unroll
    for (int t = 0; t < 8; ++t) {
      int col = t * 16 + li;
      v2f b;
      b.x = W[(k + 2 * half    ) * DH + col];
      b.y = W[(k + 2 * half + 1) * DH + col];
      acc[t] = __builtin_amdgcn_wmma_f32_16x16x4_f32(
          false, a, false, b, (short)0, acc[t], false, false);
    }
  }

  int m0 = tileM * 16 + half * 8;
#pragma unroll
  for (int t = 0; t < 8; ++t) {
    int col = t * 16 + li;
#pragma unroll
    for (int i = 0; i < 8; ++i)
      C[(size_t)(m0 + i) * DH + col] = acc[t][i];
  }
}

// ---------------- seed accumulator: bias + self-loop term ----------------

__global__ __launch_bounds__(256) void k_seed(const float* __restrict__ hw,
                                              const float* __restrict__ dinv,
                                              const float* __restrict__ bias,
                                              float* __restrict__ acc) {
  int idx  = blockIdx.x * 256 + threadIdx.x;   // exactly NNODES*DH threads
  int node = idx >> 7;
  int k    = idx & 127;
  float di = dinv[node];
  acc[idx] = bias[k] + hw[idx] * (di * di);
}

// ---------------- edge scatter: acc[dst] += hw[src] * norm ----------------
// One wave per edge; 4 channels per lane (128 total), 16B coalesced gather.

__global__ __launch_bounds__(256) void k_scatter(const float* __restrict__ hw,
                                                 const int* __restrict__ src,
                                                 const int* __restrict__ dst,
                                                 const float* __restrict__ norm,
                                                 float* __restrict__ acc) {
  int e = blockIdx.x * 8 + (threadIdx.x >> 5);
  if (e >= NEDGES) return;
  int lane = threadIdx.x & 31;
  int s = src[e];
  int d = dst[e];
  float nw = norm[e];
  v4f v = *(const v4f*)(hw + (size_t)s * DH + lane * 4);
  float* out = acc + (size_t)d * DH + lane * 4;
  atomicAdd(out + 0, v.x * nw);
  atomicAdd(out + 1, v.y * nw);
  atomicAdd(out + 2, v.z * nw);
  atomicAdd(out + 3, v.w * nw);
}

// ---------------- PReLU ----------------

__global__ __launch_bounds__(256) void k_prelu(float* __restrict__ acc,
                                               const float* __restrict__ a) {
  int idx = blockIdx.x * 256 + threadIdx.x;
  int k   = idx & 127;
  float h = acc[idx];
  acc[idx] = (h >= 0.0f) ? h : a[k] * h;
}

__global__ __launch_bounds__(256) void k_prelu_out(const float* __restrict__ acc,
                                                   const float* __restrict__ a,
                                                   float* __restrict__ out,
                                                   int s) {
  int idx  = blockIdx.x * 256 + threadIdx.x;
  int node = idx >> 7;
  int k    = idx & 127;
  float h  = acc[idx];
  h = (h >= 0.0f) ? h : a[k] * h;
  out[((size_t)node * NSCALES + s) * DH + k] = h;  // [N, S, H]
}

// ---------------- host orchestration ----------------

extern "C" void kernel_launch(void* const* d_in, const int* in_sizes, int n_in,
                              void* d_out, int out_size, void* d_ws, size_t ws_size,
                              hipStream_t stream) {
  (void)in_sizes; (void)n_in; (void)out_size; (void)ws_size;
  const float* x  = (const float*)d_in[0];
  const int*   ei = (const int*)  d_in[1];   // [S, 2, E]
  const float* ew = (const float*)d_in[2];   // [S, E]
  const float* W1 = (const float*)d_in[3];   // [S, 128, 128]
  const float* b1 = (const float*)d_in[4];   // [S, 128]
  const float* W2 = (const float*)d_in[5];
  const float* b2 = (const float*)d_in[6];
  const float* pa = (const float*)d_in[7];   // [S, 128]
  float* out = (float*)d_out;

  float* ws   = (float*)d_ws;
  float* dinv = ws;                                // N
  float* norm = dinv + NNODES;                     // E
  float* hw   = norm + NEDGES;                     // N*128
  float* acc  = hw + (size_t)NNODES * DH;          // N*128

  const int NB_NODE = (NNODES + 255) / 256;
  const int NB_EDGE = (NEDGES + 255) / 256;
  const int NB_NC   = (NNODES * DH) / 256;         // exact
  const int NB_SCAT = NEDGES / 8;                  // exact
  const int NB_GEMM = (MTILES + 7) / 8;

  for (int s = 0; s < NSCALES; ++s) {
    const int*   src = ei + (size_t)s * 2 * NEDGES;
    const int*   dst = src + NEDGES;
    const float* w   = ew + (size_t)s * NEDGES;

    k_deg_init <<<NB_NODE, 256, 0, stream>>>(dinv);
    k_deg_accum<<<NB_EDGE, 256, 0, stream>>>(dst, w, dinv);
    k_dinv     <<<NB_NODE, 256, 0, stream>>>(dinv);
    k_norm     <<<NB_EDGE, 256, 0, stream>>>(src, dst, w, dinv, norm);

    // layer 1: h1 = prelu(scatter(x@W1) + b1)
    k_gemm   <<<NB_GEMM, 256, 0, stream>>>(x, W1 + (size_t)s * DH * DH, hw);
    k_seed   <<<NB_NC,   256, 0, stream>>>(hw, dinv, b1 + s * DH, acc);
    k_scatter<<<NB_SCAT, 256, 0, stream>>>(hw, src, dst, norm, acc);
    k_prelu  <<<NB_NC,   256, 0, stream>>>(acc, pa + s * DH);

    // layer 2: out[:,s,:] = prelu(scatter(h1@W2) + b2)
    k_gemm     <<<NB_GEMM, 256, 0, stream>>>(acc, W2 + (size_t)s * DH * DH, hw);
    k_seed     <<<NB_NC,   256, 0, stream>>>(hw, dinv, b2 + s * DH, acc);
    k_scatter  <<<NB_SCAT, 256, 0, stream>>>(hw, src, dst, norm, acc);
    k_prelu_out<<<NB_NC,   256, 0, stream>>>(acc, pa + s * DH, out, s);
  }
}